// CausalAttention_47407849013605
// MI455X (gfx1250) — compile-verified
//
#include <hip/hip_runtime.h>
#include <hip/hip_bf16.h>
#include <cstdint>

typedef __attribute__((ext_vector_type(2))) float v2f;
typedef __attribute__((ext_vector_type(4))) float v4f;
typedef __attribute__((ext_vector_type(8))) float v8f;
typedef __attribute__((ext_vector_type(4))) int   v4i;

#define B_ 8
#define T_ 16
#define C_ 2048
#define H_ 16
#define HD_ 128
#define MAX_LEN_ 4096
#define CACHE_LEN_ 2048
#define NEW_LEN_ (CACHE_LEN_ + T_)          // 2064
#define NKT_ ((NEW_LEN_ + 15) / 16)         // 129 key tiles
#define QKV_N_ (3 * C_)                     // 6144
#define M_ (B_ * T_)                        // 128
#define SPLITS_ 8
#define TILES_PER_SPLIT_ 17                 // 8*17 = 136 >= 129

__device__ inline v8f wmma4(v2f a, v2f b, v8f c) {
  // V_WMMA_F32_16X16X4_F32 : D = A(16x4) * B(4x16) + C(16x16)
  return __builtin_amdgcn_wmma_f32_16x16x4_f32(false, a, false, b, (short)0, c,
                                               false, false);
}

__device__ inline float redmax16(float v) {
#pragma unroll
  for (int o = 1; o < 16; o <<= 1) v = fmaxf(v, __shfl_xor(v, o, 16));
  return v;
}
__device__ inline float redsum16(float v) {
#pragma unroll
  for (int o = 1; o < 16; o <<= 1) v += __shfl_xor(v, o, 16);
  return v;
}

// ---------------------------------------------------------------------------
// f32 WMMA GEMM  C[M,N] = A[M,K] * B[K,N]
// block = 128 threads (4 waves). grid = (M/16, N/64).
// ---------------------------------------------------------------------------
__global__ __launch_bounds__(128) void gemm_f32_wmma(
    const float* __restrict__ A, const float* __restrict__ Bm,
    float* __restrict__ Cm, int M, int N, int K) {
  __shared__ float lsA[16 * 34];
  __shared__ float lsB[32 * 64];

  const int tid  = threadIdx.x;
  const int w    = tid >> 5;
  const int lane = tid & 31;
  const int col  = lane & 15;
  const int half = lane >> 4;
  const int m0   = blockIdx.x * 16;
  const int n0   = blockIdx.y * 64;

  v8f acc = {};

  for (int k0 = 0; k0 < K; k0 += 32) {
    __syncthreads();
    {
      const int r  = tid >> 3;
      const int c4 = (tid & 7) * 4;
      v4f va = *(const v4f*)(A + (size_t)(m0 + r) * K + k0 + c4);
      lsA[r * 34 + c4 + 0] = va.x;
      lsA[r * 34 + c4 + 1] = va.y;
      lsA[r * 34 + c4 + 2] = va.z;
      lsA[r * 34 + c4 + 3] = va.w;
    }
#pragma unroll
    for (int rr = 0; rr < 4; ++rr) {
      const int r  = (tid >> 4) * 4 + rr;
      const int c4 = (tid & 15) * 4;
      *(v4f*)&lsB[r * 64 + c4] =
          *(const v4f*)(Bm + (size_t)(k0 + r) * N + n0 + c4);
    }
    __syncthreads();

    const int nl = w * 16;
#pragma unroll
    for (int kk = 0; kk < 32; kk += 4) {
      v2f a = *(const v2f*)&lsA[col * 34 + kk + 2 * half];
      v2f b;
      b.x = lsB[(kk + 2 * half + 0) * 64 + nl + col];
      b.y = lsB[(kk + 2 * half + 1) * 64 + nl + col];
      acc = wmma4(a, b, acc);
    }
  }

#pragma unroll
  for (int i = 0; i < 8; ++i)
    Cm[(size_t)(m0 + i + 8 * half) * N + n0 + w * 16 + col] = acc[i];
}

// ---------------------------------------------------------------------------
// RoPE on q and k inside the qkv workspace (in place).
// ---------------------------------------------------------------------------
__global__ void rope_kernel(float* __restrict__ qkv,
                            const float* __restrict__ cos_tab,
                            const float* __restrict__ sin_tab) {
  const int tid = blockIdx.x * blockDim.x + threadIdx.x;
  if (tid >= B_ * T_ * H_ * 64) return;
  const int j = tid & 63;
  const int h = (tid >> 6) & 15;
  const int t = (tid >> 10) & 15;
  const int b = tid >> 14;
  const int pos = CACHE_LEN_ + t;
  const float c = cos_tab[pos * 64 + j];
  const float s = sin_tab[pos * 64 + j];
  const size_t base = (size_t)(b * T_ + t) * QKV_N_ + h * HD_ + 2 * j;

  v2f q = *(v2f*)(qkv + base);
  v2f k = *(v2f*)(qkv + base + C_);
  v2f qo, ko;
  qo.x = q.x * c - q.y * s;  qo.y = q.y * c + q.x * s;
  ko.x = k.x * c - k.y * s;  ko.y = k.y * c + k.x * s;
  *(v2f*)(qkv + base)      = qo;
  *(v2f*)(qkv + base + C_) = ko;
}

// ---------------------------------------------------------------------------
// Stream k_buf / v_buf -> d_out with the T new rows injected (NT b128).
// ---------------------------------------------------------------------------
__global__ void inject_copy(const v4f* __restrict__ src, v4f* __restrict__ dst,
                            const float* __restrict__ qkv, int col_off) {
  const int NV4 = B_ * H_ * MAX_LEN_ * (HD_ / 4);
  for (int idx = blockIdx.x * blockDim.x + threadIdx.x; idx < NV4;
       idx += gridDim.x * blockDim.x) {
    const int d4  = idx & 31;
    const int row = idx >> 5;
    const int pos = row & (MAX_LEN_ - 1);
    const int bh  = row >> 12;
    v4f v;
    if (pos >= CACHE_LEN_ && pos < NEW_LEN_) {
      const int t = pos - CACHE_LEN_;
      const int h = bh & 15, b = bh >> 4;
      v = *(const v4f*)(qkv + (size_t)(b * T_ + t) * QKV_N_ + col_off +
                        h * HD_ + d4 * 4);
    } else {
      v = __builtin_nontemporal_load(src + idx);
    }
    __builtin_nontemporal_store(v, dst + idx);
  }
}

// ---------------------------------------------------------------------------
// Flash-decode attention.  grid = (B*H, SPLITS_); block = 128 (4 waves).
// Each block covers TILES_PER_SPLIT_ key tiles and emits one unnormalized
// partial (m, l, 16x128 acc) per (bh, split).  V tiles staged with
// GLOBAL_LOAD_ASYNC_TO_LDS_B128 (ASYNCcnt) so the DMA overlaps q*K^T WMMAs.
// ---------------------------------------------------------------------------
__global__ __launch_bounds__(128) void attn_split_kernel(
    const float* __restrict__ qkv, const float* __restrict__ kbuf,
    const float* __restrict__ vbuf, float* __restrict__ part_acc,
    float* __restrict__ part_m, float* __restrict__ part_l) {
  __shared__ float lsV[4][16 * 128];
  __shared__ float lsP[4][16 * 18];
  __shared__ float sm[4][16];
  __shared__ float sl[4][16];

  const int tid   = threadIdx.x;
  const int w     = tid >> 5;
  const int lane  = tid & 31;
  const int col   = lane & 15;
  const int half  = lane >> 4;
  const int bh    = blockIdx.x;
  const int split = blockIdx.y;
  const int b     = bh >> 4;
  const int h     = bh & 15;

  // q fragments (scaled), A-layout: row = col, K pair chosen by half
  const float scale = 0.088388347648318447f;  // 1/sqrt(128)
  const float* qrow = qkv + (size_t)(b * T_ + col) * QKV_N_ + h * HD_;
  v2f qa[32];
#pragma unroll
  for (int f = 0; f < 32; ++f) {
    v2f t = *(const v2f*)(qrow + f * 4 + 2 * half);
    qa[f].x = t.x * scale;
    qa[f].y = t.y * scale;
  }

  v8f acc[8];
#pragma unroll
  for (int n = 0; n < 8; ++n) acc[n] = (v8f){};
  float mrow[8], lrow[8];
#pragma unroll
  for (int i = 0; i < 8; ++i) { mrow[i] = -3.0e38f; lrow[i] = 0.f; }

  float* lv = lsV[w];
  float* lp = lsP[w];

  for (int idx = w; idx < TILES_PER_SPLIT_; idx += 4) {
    const int kt = split * TILES_PER_SPLIT_ + idx;   // wave-uniform
    if (kt < NKT_) {
      // ---- stage V tile 16x128 into this wave's LDS slab (async DMA) ----
      const float* vbase = vbuf + ((size_t)bh * MAX_LEN_ + kt * 16) * HD_;
#if __has_builtin(__builtin_amdgcn_global_load_async_to_lds_b128)
#pragma unroll
      for (int r = 0; r < 16; ++r) {
        __builtin_amdgcn_global_load_async_to_lds_b128(
            (__attribute__((address_space(1))) v4i*)(uintptr_t)(
                vbase + r * 128 + lane * 4),
            (__attribute__((address_space(3))) v4i*)(uint32_t)(uintptr_t)(
                &lv[r * 128 + lane * 4]),
            0, 0);
      }
#else
#pragma unroll
      for (int r = 0; r < 16; ++r)
        *(v4f*)&lv[r * 128 + lane * 4] =
            *(const v4f*)(vbase + r * 128 + lane * 4);
#endif

      // ---- S = q * K^T (B-frags contiguous in head-dim) ----
      v8f s = {};
      const float* kbase = kbuf + ((size_t)bh * MAX_LEN_ + kt * 16 + col) * HD_;
#pragma unroll
      for (int f = 0; f < 32; ++f) {
        v2f bfrag = *(const v2f*)(kbase + f * 4 + 2 * half);
        s = wmma4(qa[f], bfrag, s);
      }

      // causal mask: only the final tile (keys 2048..2063) is partial
      if (kt == NKT_ - 1) {
#pragma unroll
        for (int i = 0; i < 8; ++i)
          if (col > i + 8 * half) s[i] = -3.0e38f;
      }

      // ---- online softmax (row r = i + 8*half across the 16 col-lanes) ----
#pragma unroll
      for (int i = 0; i < 8; ++i) {
        const float tmax = redmax16(s[i]);
        const float nm = fmaxf(mrow[i], tmax);
        const float alpha = __expf(mrow[i] - nm);
        mrow[i] = nm;
        lrow[i] *= alpha;
#pragma unroll
        for (int n = 0; n < 8; ++n) acc[n][i] *= alpha;
        const float p = __expf(s[i] - nm);
        lrow[i] += redsum16(p);
        lp[(i + 8 * half) * 18 + col] = p;
      }

      // async V landing + all lanes' P stores must complete before reads
      asm volatile("s_wait_asynccnt 0" ::: "memory");
      asm volatile("s_wait_dscnt 0" ::: "memory");

      // ---- P * V ----
      v2f pa[4];
#pragma unroll
      for (int kb4 = 0; kb4 < 4; ++kb4)
        pa[kb4] = *(const v2f*)&lp[col * 18 + kb4 * 4 + 2 * half];
#pragma unroll
      for (int n = 0; n < 8; ++n) {
#pragma unroll
        for (int kb4 = 0; kb4 < 4; ++kb4) {
          const int kb = kb4 * 4;
          v2f bv;
          bv.x = lv[(kb + 2 * half + 0) * 128 + n * 16 + col];
          bv.y = lv[(kb + 2 * half + 1) * 128 + n * 16 + col];
          acc[n] = wmma4(pa[kb4], bv, acc[n]);
        }
      }
    }
  }

  // ---- intra-block merge of the 4 wave partials (reuse lsV slab) ----
#pragma unroll
  for (int n = 0; n < 8; ++n)
#pragma unroll
    for (int i = 0; i < 8; ++i)
      lv[(i + 8 * half) * 128 + n * 16 + col] = acc[n][i];
  if (col == 0) {
#pragma unroll
    for (int i = 0; i < 8; ++i) {
      sm[w][i + 8 * half] = mrow[i];
      sl[w][i + 8 * half] = lrow[i];
    }
  }
  __syncthreads();

  const int c = tid;  // head-dim column
  const size_t pbase = (size_t)(bh * SPLITS_ + split) * 16;
  for (int r = 0; r < 16; ++r) {
    float M = sm[0][r];
#pragma unroll
    for (int w2 = 1; w2 < 4; ++w2) M = fmaxf(M, sm[w2][r]);
    float L = 0.f, val = 0.f;
#pragma unroll
    for (int w2 = 0; w2 < 4; ++w2) {
      const float e = __expf(sm[w2][r] - M);
      L += sl[w2][r] * e;
      val += lsV[w2][r * 128 + c] * e;
    }
    part_acc[(pbase + r) * 128 + c] = val;   // unnormalized
    if (c == r) { part_m[pbase + r] = M; part_l[pbase + r] = L; }
  }
}

// ---------------------------------------------------------------------------
// Merge the SPLITS_ partials per (b,h) with log-sum-exp.
// ---------------------------------------------------------------------------
__global__ __launch_bounds__(128) void attn_merge_kernel(
    const float* __restrict__ part_acc, const float* __restrict__ part_m,
    const float* __restrict__ part_l, float* __restrict__ attn_out) {
  const int bh = blockIdx.x;
  const int b = bh >> 4, h = bh & 15;
  const int c = threadIdx.x;
  for (int r = 0; r < 16; ++r) {
    float M = -3.0e38f;
#pragma unroll
    for (int s = 0; s < SPLITS_; ++s)
      M = fmaxf(M, part_m[(bh * SPLITS_ + s) * 16 + r]);
    float L = 0.f, val = 0.f;
#pragma unroll
    for (int s = 0; s < SPLITS_; ++s) {
      const size_t pb = (size_t)(bh * SPLITS_ + s) * 16 + r;
      const float e = __expf(part_m[pb] - M);
      L += part_l[pb] * e;
      val += part_acc[pb * 128 + c] * e;
    }
    attn_out[(size_t)(b * T_ + r) * C_ + h * HD_ + c] = val / L;
  }
}

// ---------------------------------------------------------------------------
extern "C" void kernel_launch(void* const* d_in, const int* in_sizes, int n_in,
                              void* d_out, int out_size, void* d_ws,
                              size_t ws_size, hipStream_t stream) {
  const float* x       = (const float*)d_in[0];
  const float* k_buf   = (const float*)d_in[1];
  const float* v_buf   = (const float*)d_in[2];
  const float* W_qkv   = (const float*)d_in[3];
  const float* W_out   = (const float*)d_in[4];
  const float* cos_tab = (const float*)d_in[5];
  const float* sin_tab = (const float*)d_in[6];

  float* out      = (float*)d_out;                    // (B,T,C)
  float* out_kbuf = out + (size_t)M_ * C_;            // (B,H,MAX_LEN,HD)
  float* out_vbuf = out_kbuf + (size_t)B_ * H_ * MAX_LEN_ * HD_;

  float* ws_qkv   = (float*)d_ws;                               // 128*6144
  float* ws_attn  = ws_qkv + (size_t)M_ * QKV_N_;               // 128*2048
  float* ws_pacc  = ws_attn + (size_t)M_ * C_;                  // 128*8*16*128
  float* ws_pm    = ws_pacc + (size_t)B_ * H_ * SPLITS_ * 16 * HD_;
  float* ws_pl    = ws_pm + (size_t)B_ * H_ * SPLITS_ * 16;

  // 1) qkv = x @ W_qkv
  gemm_f32_wmma<<<dim3(M_ / 16, QKV_N_ / 64), 128, 0, stream>>>(
      x, W_qkv, ws_qkv, M_, QKV_N_, C_);

  // 2) RoPE on q,k
  rope_kernel<<<(B_ * T_ * H_ * 64 + 255) / 256, 256, 0, stream>>>(
      ws_qkv, cos_tab, sin_tab);

  // 3) cache pass-through with new rows injected (NT streams)
  inject_copy<<<4096, 256, 0, stream>>>((const v4f*)k_buf, (v4f*)out_kbuf,
                                        ws_qkv, C_);
  inject_copy<<<4096, 256, 0, stream>>>((const v4f*)v_buf, (v4f*)out_vbuf,
                                        ws_qkv, 2 * C_);

  // 4) flash-decode attention + merge
  attn_split_kernel<<<dim3(B_ * H_, SPLITS_), 128, 0, stream>>>(
      ws_qkv, out_kbuf, out_vbuf, ws_pacc, ws_pm, ws_pl);
  attn_merge_kernel<<<B_ * H_, 128, 0, stream>>>(ws_pacc, ws_pm, ws_pl,
                                                 ws_attn);

  // 5) out = attn @ W_out
  gemm_f32_wmma<<<dim3(M_ / 16, C_ / 64), 128, 0, stream>>>(
      ws_attn, W_out, out, M_, C_, C_);
}